// OTEERE_72327249264650
// MI455X (gfx1250) — compile-verified
//
#include <hip/hip_runtime.h>
#include <cstdint>
#include <cstddef>

// Problem constants (match reference)
#define B_  16
#define S_  4096
#define H_  1024
#define N_  2048
#define DW_ 50
#define OUTW (H_ + DW_)      // 1074 floats per output row
#define LWPAD 64             // padded width for last_w workspace

// ---------- order-preserving f32 <-> u32 mapping for atomic max ----------
__device__ __forceinline__ unsigned f2ord(float f) {
  unsigned u = __float_as_uint(f);
  return (u & 0x80000000u) ? ~u : (u | 0x80000000u);
}
__device__ __forceinline__ float ord2f(unsigned m) {
  unsigned u = (m & 0x80000000u) ? (m & 0x7FFFFFFFu) : ~m;
  return __uint_as_float(u);
}
#define ORD_NEG_INF 0x007FFFFFu   // f2ord(-inf)

// ---------- CDNA5 async global->LDS copy (ASYNCcnt-tracked) ----------
// Each issue copies 16B per active lane: from the lane's global address GA+OFF to its LDS address LDSA+OFF
#define ASYNC_CP_B128(LDSA, GA, OFF)                                          \
  asm volatile("global_load_async_to_lds_b128 %0, %1, off offset:" #OFF       \
               :: "v"(LDSA), "v"(GA) : "memory")
#define WAIT_ASYNC_8() asm volatile("s_wait_asynccnt 0x8" ::: "memory")
#define WAIT_ASYNC_0() asm volatile("s_wait_asynccnt 0x0" ::: "memory")

// Stage one 4KB context row (1024 f32) into LDS: 8 wave-wide b128 async issues.
__device__ __forceinline__ void stage_row(unsigned ldsBase, const float* gp, int lane) {
  unsigned l = ldsBase + (unsigned)lane * 16u;
  unsigned long long g =
      (unsigned long long)(uintptr_t)gp + (unsigned long long)lane * 16ull;
  ASYNC_CP_B128(l, g, 0);
  ASYNC_CP_B128(l, g, 512);
  ASYNC_CP_B128(l, g, 1024);
  ASYNC_CP_B128(l, g, 1536);
  ASYNC_CP_B128(l, g, 2048);
  ASYNC_CP_B128(l, g, 2560);
  ASYNC_CP_B128(l, g, 3072);
  ASYNC_CP_B128(l, g, 3584);
}

// ---------------- kernel 1: init workspace ----------------
__global__ void k_init(int* __restrict__ starts, unsigned* __restrict__ lc,
                       unsigned* __restrict__ lw) {
  int i = blockIdx.x * 256 + threadIdx.x;
  if (i < B_ * (N_ + 1)) starts[i] = S_;
  if (i < B_ * H_)       lc[i] = ORD_NEG_INF;
  if (i < B_ * LWPAD)    lw[i] = ORD_NEG_INF;
}

// ---------------- kernel 2: span starts from sorted seg_ids ----------------
// seg_ids[b] is sorted ascending; each value's first occurrence is its span start.
__global__ void k_starts(const int* __restrict__ seg, int* __restrict__ starts) {
  int b = blockIdx.y;
  int s = blockIdx.x * 256 + threadIdx.x;
  int v = seg[(size_t)b * S_ + s];
  if (s == 0) {
    starts[b * (N_ + 1) + 0] = 0;
  } else if (seg[(size_t)b * S_ + s - 1] != v) {
    starts[b * (N_ + 1) + v] = s;
  }
}

// ---------------- kernel 3: last_ctx[b,h] = max over rows 0..ns-2 ----------------
// 256 lanes x float4 spans the full H=1024 row; b128 loads for the ~100MB stream.
__global__ void k_lastctx(const float* __restrict__ ctx, const int* __restrict__ nsv,
                          unsigned* __restrict__ lc) {
  int b    = blockIdx.z;
  int col4 = threadIdx.x;                    // float4 index within row
  int M    = nsv[b] - 1;                     // rows 0..ns-2
  const float4* base = (const float4*)(ctx + (size_t)b * S_ * H_) + col4;
  float4 acc = make_float4(-__builtin_inff(), -__builtin_inff(),
                           -__builtin_inff(), -__builtin_inff());
  int stride = gridDim.y;
  for (int r = blockIdx.y; r < M; r += stride) {
    __builtin_prefetch(base + (size_t)(r + 4 * stride) * (H_ / 4), 0, 0); // global_prefetch_b8
    float4 v = base[(size_t)r * (H_ / 4)];
    acc.x = fmaxf(acc.x, v.x);
    acc.y = fmaxf(acc.y, v.y);
    acc.z = fmaxf(acc.z, v.z);
    acc.w = fmaxf(acc.w, v.w);
  }
  unsigned* p = &lc[b * H_ + 4 * col4];
  atomicMax(p + 0, f2ord(acc.x));            // global_atomic_max_u32
  atomicMax(p + 1, f2ord(acc.y));
  atomicMax(p + 2, f2ord(acc.z));
  atomicMax(p + 3, f2ord(acc.w));
}

// ---------------- kernel 4: last_w[b,c] = max over word_emb rows 0..ns-1 ----------------
__global__ void k_lastw(const float* __restrict__ wemb, const int* __restrict__ nsv,
                        unsigned* __restrict__ lw) {
  int b = blockIdx.z;
  int c = threadIdx.x;
  if (c >= DW_) return;
  int nsb = nsv[b];
  float acc = -__builtin_inff();
  for (int r = blockIdx.y; r < nsb; r += gridDim.y)
    acc = fmaxf(acc, wemb[((size_t)b * N_ + r) * DW_ + c]);
  atomicMax(&lw[b * LWPAD + c], f2ord(acc));
}

// ---------------- kernel 5: main fused output ----------------
// 1 wave32 per word token; span rows double-buffered through LDS via async copies.
__global__ void __launch_bounds__(128) k_main(
    const float* __restrict__ ctx, const float* __restrict__ wemb,
    const int* __restrict__ nsv, const int* __restrict__ starts,
    const unsigned* __restrict__ lc, const unsigned* __restrict__ lw,
    float* __restrict__ out) {
  __shared__ __align__(16) float smem[4][2][H_];   // 4 waves x 2 row buffers x 4KB

  const int b    = blockIdx.y;
  const int wid  = threadIdx.x >> 5;
  const int lane = threadIdx.x & 31;
  const int n    = blockIdx.x * 4 + wid;
  const int nsb  = nsv[b];

  float* outrow = out + ((size_t)b * N_ + n) * OUTW;   // 8B-aligned rows
  float2* o2 = (float2*)outrow;

  if (n >= nsb) {                       // padding word -> zero vector
    const float2 z = make_float2(0.f, 0.f);
    for (int j = lane; j < OUTW / 2; j += 32) o2[j] = z;
    return;
  }

  if (n == nsb - 1) {                   // special last word token
    const unsigned* lcb = lc + b * H_;
    for (int j = lane; j < H_ / 2; j += 32)
      o2[j] = make_float2(ord2f(lcb[2 * j]), ord2f(lcb[2 * j + 1]));
    for (int j = lane; j < DW_; j += 32)
      outrow[H_ + j] = ord2f(lw[b * LWPAD + j]);
    return;
  }

  // normal word: max over contiguous subword span [lo, hi)
  const int lo = starts[b * (N_ + 1) + n];
  const int hi = starts[b * (N_ + 1) + n + 1];

  float4 acc[8];
#pragma unroll
  for (int k = 0; k < 8; ++k)
    acc[k] = make_float4(-__builtin_inff(), -__builtin_inff(),
                         -__builtin_inff(), -__builtin_inff());

  if (lo < hi) {
    const float* cb = ctx + ((size_t)b * S_ + lo) * H_;   // 16B-aligned rows
    const unsigned ldsA = (unsigned)(uintptr_t)&smem[wid][0][0];
    const unsigned ldsB = (unsigned)(uintptr_t)&smem[wid][1][0];
    stage_row(ldsA, cb, lane);
    for (int r = lo; r < hi; ++r) {
      const int cur = (r - lo) & 1;
      if (r + 1 < hi) {
        stage_row(cur ? ldsA : ldsB, cb + (size_t)(r + 1 - lo) * H_, lane);
        WAIT_ASYNC_8();                 // in-order: prev row's 8 copies done
      } else {
        WAIT_ASYNC_0();                 // final row fully landed
      }
      const float4* s4 = (const float4*)&smem[wid][cur][0];
#pragma unroll
      for (int k = 0; k < 8; ++k) {     // ds_load_b128: 1024 floats / 32 lanes
        float4 v = s4[lane + 32 * k];
        acc[k].x = fmaxf(acc[k].x, v.x);
        acc[k].y = fmaxf(acc[k].y, v.y);
        acc[k].z = fmaxf(acc[k].z, v.z);
        acc[k].w = fmaxf(acc[k].w, v.w);
      }
    }
  }

  // ctx part (cols 0..1023) — b64 stores (rows only 8B-aligned)
#pragma unroll
  for (int k = 0; k < 8; ++k) {
    const int i4 = lane + 32 * k;
    o2[2 * i4]     = make_float2(acc[k].x, acc[k].y);
    o2[2 * i4 + 1] = make_float2(acc[k].z, acc[k].w);
  }
  // word-emb part (cols 1024..1073)
  for (int j = lane; j < DW_; j += 32)
    outrow[H_ + j] = wemb[((size_t)b * N_ + n) * DW_ + j];
}

// ---------------- host launcher ----------------
extern "C" void kernel_launch(void* const* d_in, const int* in_sizes, int n_in,
                              void* d_out, int out_size, void* d_ws, size_t ws_size,
                              hipStream_t stream) {
  (void)in_sizes; (void)n_in; (void)out_size; (void)ws_size;
  const float* ctx  = (const float*)d_in[0];   // [B,S,H] f32
  const float* wemb = (const float*)d_in[1];   // [B,N,DW] f32
  const int*   seg  = (const int*)d_in[2];     // [B,S] i32 (sorted per batch)
  const int*   nsv  = (const int*)d_in[3];     // [B] i32
  float* out = (float*)d_out;                  // [B,N,H+DW] f32

  // workspace layout
  int*      starts = (int*)d_ws;                          // B*(N+1)
  unsigned* lc     = (unsigned*)(starts + B_ * (N_ + 1)); // B*H
  unsigned* lw     = lc + B_ * H_;                        // B*LWPAD

  k_init<<<dim3((B_ * (N_ + 1) + 255) / 256), 256, 0, stream>>>(starts, lc, lw);
  k_starts<<<dim3(S_ / 256, B_), 256, 0, stream>>>(seg, starts);
  k_lastctx<<<dim3(1, 32, B_), 256, 0, stream>>>(ctx, nsv, lc);
  k_lastw<<<dim3(1, 8, B_), 64, 0, stream>>>(wemb, nsv, lw);
  k_main<<<dim3(N_ / 4, B_), 128, 0, stream>>>(ctx, wemb, nsv, starts, lc, lw, out);
}